// GTCRLoss_20658792694159
// MI455X (gfx1250) — compile-verified
//
#include <hip/hip_runtime.h>

// GTCRLoss: out = -0.5 * mean_i( log1p(lmbd * ||z_i||^2) ), z: [8192,4096] f32.
// Bandwidth-bound (128 MiB @ 23.3 TB/s ~ 5.6us). Row sum-of-squares done via
// V_WMMA_F32_16X16X4_F32 with a ones B-matrix: D[m][n] += sum_k A[m][k]*1.

typedef __attribute__((ext_vector_type(2))) float v2f;
typedef __attribute__((ext_vector_type(4))) float v4f;
typedef __attribute__((ext_vector_type(8))) float v8f;

#define D_DIM 4096
#define B_DIM 8192
#define WAVES_PER_BLOCK 8
#define KSLICE (D_DIM / WAVES_PER_BLOCK) // 512 columns per wave

__global__ __launch_bounds__(256) void gtcr_rowtile_kernel(
    const float* __restrict__ z, float* __restrict__ partial, float lmbd) {
  __shared__ float s_part[WAVES_PER_BLOCK * 16];
  __shared__ float s_log[16];

  const int tid  = threadIdx.x;
  const int wave = tid >> 5;
  const int lane = tid & 31;
  const int row  = lane & 15;          // A-matrix: lanes 0-15 and 16-31 both hold M=0..15
  const int koff = (lane >> 4) << 2;   // lane r -> cols k0..k0+3, lane r+16 -> k0+4..k0+7

  const size_t rowBase = (size_t)(blockIdx.x * 16 + row) * (size_t)D_DIM;
  const float* p = z + rowBase + (size_t)(wave * KSLICE + koff);

  v8f acc0 = {};
  v8f acc1 = {};
  const v2f ones = {1.0f, 1.0f};

  // 64 iterations: one 16B load, two chained-independent WMMA accumulators.
  #pragma unroll 4
  for (int kk = 0; kk < KSLICE; kk += 8) {
    v4f q = *(const v4f*)(p + kk);     // global_load_b128, per-lane contiguous
    v2f a0 = {q.x * q.x, q.y * q.y};   // K-chunk 0 (permuted columns: sum-invariant)
    v2f a1 = {q.z * q.z, q.w * q.w};   // K-chunk 1
    acc0 = __builtin_amdgcn_wmma_f32_16x16x4_f32(
        false, a0, false, ones, (short)0, acc0, false, false);
    acc1 = __builtin_amdgcn_wmma_f32_16x16x4_f32(
        false, a1, false, ones, (short)0, acc1, false, false);
  }

  union { v8f v; float f[8]; } acc;
  acc.v = acc0 + acc1;

  // C/D layout: c[m] lane 0 -> row m ; c[m] lane 16 -> row m+8 (every N column
  // of D holds the same row-sum, so one lane per half suffices).
  if (lane == 0) {
    #pragma unroll
    for (int m = 0; m < 8; ++m) s_part[wave * 16 + m] = acc.f[m];
  } else if (lane == 16) {
    #pragma unroll
    for (int m = 0; m < 8; ++m) s_part[wave * 16 + 8 + m] = acc.f[m];
  }
  __syncthreads();

  // Deterministic cross-wave reduction + per-row log1p.
  if (tid < 16) {
    float s = 0.0f;
    #pragma unroll
    for (int w = 0; w < WAVES_PER_BLOCK; ++w) s += s_part[w * 16 + tid];
    s_log[tid] = log1pf(lmbd * s);
  }
  __syncthreads();

  if (tid == 0) {
    float t = 0.0f;
    #pragma unroll
    for (int i = 0; i < 16; ++i) t += s_log[i];
    partial[blockIdx.x] = t;  // one partial per 16-row tile (512 total)
  }
}

__global__ __launch_bounds__(256) void gtcr_finalize_kernel(
    const float* __restrict__ partial, float* __restrict__ out) {
  __shared__ float s[256];
  const int t = threadIdx.x;
  s[t] = partial[t] + partial[t + 256];  // 512 partials -> 256
  __syncthreads();
  if (t == 0) {
    float tot = 0.0f;
    for (int i = 0; i < 256; ++i) tot += s[i];  // fixed order: deterministic
    out[0] = -0.5f * tot / (float)B_DIM;
  }
}

extern "C" void kernel_launch(void* const* d_in, const int* in_sizes, int n_in,
                              void* d_out, int out_size, void* d_ws, size_t ws_size,
                              hipStream_t stream) {
  (void)in_sizes; (void)n_in; (void)out_size; (void)ws_size;
  const float* z = (const float*)d_in[0];
  float* out = (float*)d_out;
  float* ws  = (float*)d_ws;  // 512 floats of scratch

  const float lmbd = (float)D_DIM * ((float)B_DIM * 0.001f);  // D * (B * EPSILON)

  gtcr_rowtile_kernel<<<B_DIM / 16, 256, 0, stream>>>(z, ws, lmbd);
  gtcr_finalize_kernel<<<1, 256, 0, stream>>>(ws, out);
}